// ProxyFeatureGate_52304111731212
// MI455X (gfx1250) — compile-verified
//
#include <hip/hip_runtime.h>
#include <hip/hip_bf16.h>

#define ROW_LEN   4096
#define BLOCK     256
#define K_KEEP    2048u
#define ROW_EPS   1e-6f

// ---- CDNA5 async global->LDS path (gfx1250): probe for the builtins ----
#if defined(__HIP_DEVICE_COMPILE__)
#  if __has_builtin(__builtin_amdgcn_global_load_async_to_lds_b128)
#    define HAVE_ASYNC_LDS 1
#  endif
#  if __has_builtin(__builtin_amdgcn_s_wait_asynccnt)
#    define HAVE_WAIT_ASYNC 1
#  endif
#endif
#ifndef HAVE_ASYNC_LDS
#  define HAVE_ASYNC_LDS 0
#endif
#ifndef HAVE_WAIT_ASYNC
#  define HAVE_WAIT_ASYNC 0
#endif

typedef int v4i_t __attribute__((ext_vector_type(4)));
typedef __attribute__((address_space(1))) v4i_t gv4i_t;  // global int4
typedef __attribute__((address_space(3))) v4i_t lv4i_t;  // LDS int4

__device__ __forceinline__ float gate_of(unsigned u, unsigned thr, float mu, float isig)
{
    // soft gate: sigmoid((s - mu)/sigma); hard top-k entries get exactly 1.0
    const float z    = (__uint_as_float(u) - mu) * isig;
    const float soft = 1.0f / (1.0f + __expf(-z));
    return (u >= thr) ? 1.0f : soft;
}

__global__ __launch_bounds__(BLOCK) void
proxy_gate_kernel(const float* __restrict__ token,
                  const float* __restrict__ proxy,
                  float* __restrict__ out)
{
    __shared__ __align__(16) float s_px[ROW_LEN];  // 16 KB staged proxy row
    __shared__ unsigned s_hist[256];
    __shared__ unsigned s_scan[256];
    __shared__ float    s_red[2][256];
    __shared__ unsigned s_prefix;
    __shared__ unsigned s_want;
    __shared__ float    s_mu;
    __shared__ float    s_isig;

    const unsigned tid     = threadIdx.x;
    const size_t   rowBase = (size_t)blockIdx.x * ROW_LEN;
    const float* tok_row = token + rowBase;
    const float* px_row  = proxy + rowBase;
    float*       out_row = out   + rowBase;

    // ---- Stage: token -> registers (b128), proxy -> LDS (async b128) ----
    float4 tok[4];
#pragma unroll
    for (int j = 0; j < 4; ++j) {
        const unsigned v = (unsigned)j * BLOCK + tid;     // float4 index in row
#if HAVE_ASYNC_LDS
        __builtin_amdgcn_global_load_async_to_lds_b128(
            (gv4i_t*)(px_row + 4u * v),
            (lv4i_t*)(&s_px[4u * v]),
            /*imm offset*/0, /*cpol*/0);
#else
        ((float4*)s_px)[v] = ((const float4*)px_row)[v];
#endif
        tok[j] = ((const float4*)tok_row)[v];
    }
#if HAVE_ASYNC_LDS
#  if HAVE_WAIT_ASYNC
    __builtin_amdgcn_s_wait_asynccnt(0);
#  else
    asm volatile("s_wait_asynccnt 0" ::: "memory");
#  endif
#endif
    __syncthreads();

    // ---- score = |tok*px| as IEEE bits (non-negative floats order as u32) ----
    unsigned sb[16];
    float sum = 0.0f, sq = 0.0f;
#pragma unroll
    for (int j = 0; j < 4; ++j) {
        const unsigned v = (unsigned)j * BLOCK + tid;
        const float4 p = ((const float4*)s_px)[v];
        const float4 t = tok[j];
        const float s0 = fabsf(t.x * p.x);
        const float s1 = fabsf(t.y * p.y);
        const float s2 = fabsf(t.z * p.z);
        const float s3 = fabsf(t.w * p.w);
        sb[4*j+0] = __float_as_uint(s0);
        sb[4*j+1] = __float_as_uint(s1);
        sb[4*j+2] = __float_as_uint(s2);
        sb[4*j+3] = __float_as_uint(s3);
        sum += (s0 + s1) + (s2 + s3);
        sq  += (s0*s0 + s1*s1) + (s2*s2 + s3*s3);
    }

    // ---- block reduction: mean + unbiased std ----
    s_red[0][tid] = sum;
    s_red[1][tid] = sq;
    __syncthreads();
#pragma unroll
    for (unsigned off = BLOCK / 2; off >= 1; off >>= 1) {
        if (tid < off) {
            s_red[0][tid] += s_red[0][tid + off];
            s_red[1][tid] += s_red[1][tid + off];
        }
        __syncthreads();
    }
    if (tid == 0) {
        const float S  = s_red[0][0];
        const float Q  = s_red[1][0];
        const float mu = S * (1.0f / (float)ROW_LEN);
        float var = (Q - S * mu) * (1.0f / (float)(ROW_LEN - 1)); // ddof=1
        var = fmaxf(var, 0.0f);
        const float sd = fmaxf(sqrtf(var), ROW_EPS);
        s_mu     = mu;
        s_isig   = 1.0f / sd;
        s_prefix = 0u;
        s_want   = K_KEEP;
    }
    __syncthreads();

    // ---- 4-pass radix select: bit pattern of the K_KEEP-th largest score ----
    for (int pass = 0; pass < 4; ++pass) {
        const int      shift  = 24 - 8 * pass;
        const unsigned prefix = s_prefix;                  // synced last pass
        const unsigned want   = s_want;
        const unsigned himask = (pass == 0) ? 0u : (0xFFFFFFFFu << (shift + 8));

        s_hist[tid] = 0u;
        __syncthreads();
#pragma unroll
        for (int i = 0; i < 16; ++i) {
            const unsigned u = sb[i];
            if ((u & himask) == prefix)
                atomicAdd(&s_hist[(u >> shift) & 0xFFu], 1u);
        }
        __syncthreads();

        // inclusive suffix-sum over 256 bins (Hillis-Steele)
        s_scan[tid] = s_hist[tid];
        __syncthreads();
        for (unsigned off = 1; off < 256; off <<= 1) {
            const unsigned mine = s_scan[tid];
            const unsigned plus = (tid + off < 256) ? s_scan[tid + off] : 0u;
            __syncthreads();
            s_scan[tid] = mine + plus;
            __syncthreads();
        }
        const unsigned ge = s_scan[tid];                       // count digit >= tid
        const unsigned gt = (tid < 255) ? s_scan[tid + 1] : 0u;// count digit >  tid
        if (ge >= want && gt < want) {                         // exactly one thread
            s_prefix = prefix | (tid << shift);
            s_want   = want - gt;
        }
        __syncthreads();
    }

    // ---- gate + output (b128 stores) ----
    const unsigned thr  = s_prefix;
    const float    mu   = s_mu;
    const float    isig = s_isig;
#pragma unroll
    for (int j = 0; j < 4; ++j) {
        const unsigned v = (unsigned)j * BLOCK + tid;
        const float4 t = tok[j];
        float4 o;
        o.x = t.x * gate_of(sb[4*j+0], thr, mu, isig);
        o.y = t.y * gate_of(sb[4*j+1], thr, mu, isig);
        o.z = t.z * gate_of(sb[4*j+2], thr, mu, isig);
        o.w = t.w * gate_of(sb[4*j+3], thr, mu, isig);
        ((float4*)out_row)[v] = o;
    }
}

extern "C" void kernel_launch(void* const* d_in, const int* in_sizes, int n_in,
                              void* d_out, int out_size, void* d_ws, size_t ws_size,
                              hipStream_t stream) {
    (void)n_in; (void)out_size; (void)d_ws; (void)ws_size;
    const float* token = (const float*)d_in[0];
    const float* proxy = (const float*)d_in[1];
    float*       out   = (float*)d_out;
    const int rows = in_sizes[0] / ROW_LEN;   // B = 8192 for the reference shapes
    proxy_gate_kernel<<<dim3(rows), dim3(BLOCK), 0, stream>>>(token, proxy, out);
}